// PtModule_76166950027621
// MI455X (gfx1250) — compile-verified
//
#include <hip/hip_runtime.h>
#include <hip/hip_bf16.h>

// Elementwise relu((x + 2) * 3 - 4) over 32*2048*1024 fp32 elements.
// Pure streaming kernel: 256 MiB in + 256 MiB out, HBM-bound (~22.5 us at
// 23.3 TB/s). Strategy: float4 (b128) global loads/stores with non-temporal
// cache hints, 4 outstanding b128 loads per thread, wave32-friendly blocks.

typedef __attribute__((ext_vector_type(4))) float v4f;

__device__ __forceinline__ v4f affine_relu4(v4f x) {
  v4f r;
#pragma unroll
  for (int c = 0; c < 4; ++c) {
    // Keep the exact reference op order for bitwise-identical results.
    float t = (x[c] + 2.0f) * 3.0f - 4.0f;
    r[c] = fmaxf(t, 0.0f);   // v_max_num_f32
  }
  return r;
}

__global__ __launch_bounds__(256) void relu_affine_vec4(
    const v4f* __restrict__ in, v4f* __restrict__ out, long long n4) {
  const long long stride = (long long)gridDim.x * blockDim.x;
  long long i = (long long)blockIdx.x * blockDim.x + threadIdx.x;

  // Main loop: 4 independent global_load_b128 issued back-to-back (NT hint)
  // before any use -> 4 outstanding LOADcnt entries per wave to hide HBM
  // latency; stores are NT so the 512 MiB stream doesn't churn L2/WGP$.
  for (; i + 3 * stride < n4; i += 4 * stride) {
    v4f a0 = __builtin_nontemporal_load(&in[i]);
    v4f a1 = __builtin_nontemporal_load(&in[i + stride]);
    v4f a2 = __builtin_nontemporal_load(&in[i + 2 * stride]);
    v4f a3 = __builtin_nontemporal_load(&in[i + 3 * stride]);
    __builtin_nontemporal_store(affine_relu4(a0), &out[i]);
    __builtin_nontemporal_store(affine_relu4(a1), &out[i + stride]);
    __builtin_nontemporal_store(affine_relu4(a2), &out[i + 2 * stride]);
    __builtin_nontemporal_store(affine_relu4(a3), &out[i + 3 * stride]);
  }
  // Vector remainder (not hit for 64Mi elements, kept for generality).
  for (; i < n4; i += stride) {
    v4f a = __builtin_nontemporal_load(&in[i]);
    __builtin_nontemporal_store(affine_relu4(a), &out[i]);
  }
}

// Scalar tail for element counts not divisible by 4 (not hit here).
__global__ __launch_bounds__(256) void relu_affine_tail(
    const float* __restrict__ in, float* __restrict__ out,
    long long start, long long n) {
  long long i = start + (long long)blockIdx.x * blockDim.x + threadIdx.x;
  if (i < n) {
    float t = (in[i] + 2.0f) * 3.0f - 4.0f;
    out[i] = fmaxf(t, 0.0f);
  }
}

extern "C" void kernel_launch(void* const* d_in, const int* in_sizes, int n_in,
                              void* d_out, int out_size, void* d_ws, size_t ws_size,
                              hipStream_t stream) {
  (void)in_sizes; (void)n_in; (void)d_ws; (void)ws_size;
  const float* in = (const float*)d_in[0];
  float* out = (float*)d_out;

  const long long n  = (long long)out_size;   // 32*2048*1024 = 67,108,864
  const long long n4 = n / 4;                 // 16,777,216 float4's

  if (n4 > 0) {
    const int block = 256;                    // 8 wave32 per block
    // One unrolled iteration (4 float4 = 64 B) per thread.
    long long want = (n4 + 4LL * block - 1) / (4LL * block);  // 16384 blocks
    if (want < 1) want = 1;
    if (want > 0x7FFFFFFFLL) want = 0x7FFFFFFFLL;
    relu_affine_vec4<<<(int)want, block, 0, stream>>>(
        (const v4f*)in, (v4f*)out, n4);
  }

  const long long done = n4 * 4;
  if (done < n) {
    const int block = 256;
    long long tail = n - done;
    int grid = (int)((tail + block - 1) / block);
    relu_affine_tail<<<grid, block, 0, stream>>>(in, out, done, n);
  }
}